// ObjectDetector_33148557590842
// MI455X (gfx1250) — compile-verified
//
#include <hip/hip_runtime.h>
#include <hip/hip_bf16.h>

#define NB        4
#define NLVL      4
#define KSEL      512
#define TOT_ANCH  65280          // 49152 + 12288 + 3072 + 768
#define N_LVL0    49152

// ---------- helpers ----------

__device__ __forceinline__ float clipf(float x, float lo, float hi) {
    return fminf(fmaxf(x, lo), hi);
}

// order-preserving float -> uint32 map (no NaNs in our keys)
__device__ __forceinline__ unsigned f2u(float f) {
    unsigned s = __float_as_uint(f);
    return (s & 0x80000000u) ? ~s : (s | 0x80000000u);
}

// byte offset of a generic pointer to __shared__ inside the wave's LDS
// (generic LDS pointers are {aperture_hi, lds_offset}; low 32 bits = LDS addr)
__device__ __forceinline__ unsigned lds_off(const void* p) {
    return (unsigned)(unsigned long long)p;
}

// CDNA5 async global->LDS loads (gfx1250), tracked by ASYNCcnt
__device__ __forceinline__ void async_load_b32(unsigned ldsOff, const void* gptr) {
    asm volatile("global_load_async_to_lds_b32 %0, %1, off"
                 :: "v"(ldsOff), "v"((unsigned long long)gptr) : "memory");
}
__device__ __forceinline__ void async_load_b128(unsigned ldsOff, const void* gptr) {
    asm volatile("global_load_async_to_lds_b128 %0, %1, off"
                 :: "v"(ldsOff), "v"((unsigned long long)gptr) : "memory");
}
__device__ __forceinline__ void wait_asynccnt0() {
    asm volatile("s_wait_asynccnt 0" ::: "memory");
}

// block-wide sum of a per-thread u32 (1024 threads = 32 wave32's), result to all
__device__ __forceinline__ unsigned blockSum(unsigned v, unsigned* lred, int tid) {
    #pragma unroll
    for (int off = 16; off > 0; off >>= 1)
        v += (unsigned)__shfl_xor((int)v, off, 32);
    int wid = tid >> 5, lane = tid & 31;
    if (lane == 0) lred[wid] = v;
    __syncthreads();
    if (tid == 0) {
        unsigned s = 0;
        #pragma unroll
        for (int w = 0; w < 32; w++) s += lred[w];
        lred[32] = s;
    }
    __syncthreads();
    unsigned r = lred[32];
    __syncthreads();
    return r;
}

// ---------- kernel 1: per-anchor decode ----------

__global__ __launch_bounds__(256) void decode_kernel(
    const float* __restrict__ cls0, const float* __restrict__ box0, const float* __restrict__ anc0,
    const float* __restrict__ cls1, const float* __restrict__ box1, const float* __restrict__ anc1,
    const float* __restrict__ cls2, const float* __restrict__ box2, const float* __restrict__ anc2,
    const float* __restrict__ cls3, const float* __restrict__ box3, const float* __restrict__ anc3,
    const int* __restrict__ imgp,
    float4* __restrict__ ws_box, unsigned* __restrict__ ws_u,
    float* __restrict__ ws_ms, int* __restrict__ ws_mc)
{
    int t = blockIdx.x * blockDim.x + threadIdx.x;
    if (t >= NB * TOT_ANCH) return;
    int b = t / TOT_ANCH;
    int r = t - b * TOT_ANCH;

    int lvl, i, H;
    const float *cls, *box, *anc;
    if (r < 49152)      { lvl = 0; i = r;         cls = cls0; box = box0; anc = anc0; H = 128; }
    else if (r < 61440) { lvl = 1; i = r - 49152; cls = cls1; box = box1; anc = anc1; H = 64; }
    else if (r < 64512) { lvl = 2; i = r - 61440; cls = cls2; box = box2; anc = anc2; H = 32; }
    else                { lvl = 3; i = r - 64512; cls = cls3; box = box3; anc = anc3; H = 16; }

    int HH  = H * H;
    int a   = i / HH;
    int rem = i - a * HH;
    float img = (float)(*imgp);

    // cls layout (B, NA=3, NC=3, H, H)
    int cbase = ((b * 3 + a) * 3) * HH + rem;
    float s0 = 1.f / (1.f + expf(-cls[cbase]));
    float s1 = 1.f / (1.f + expf(-cls[cbase + HH]));
    float s2 = 1.f / (1.f + expf(-cls[cbase + 2 * HH]));
    float ms = s0; int mc = 0;
    if (s1 > ms) { ms = s1; mc = 1; }
    if (s2 > ms) { ms = s2; mc = 2; }

    // box layout (B, NA=3, 4, H, H)
    int bbase = ((b * 3 + a) * 4) * HH + rem;
    float d0 = box[bbase];
    float d1 = box[bbase + HH];
    float d2 = box[bbase + 2 * HH];
    float d3 = box[bbase + 3 * HH];

    float4 av = ((const float4*)anc)[i];
    float aw  = (av.z - av.x) / img;
    float ah  = (av.w - av.y) / img;
    float acx = (av.x + av.z) * 0.5f / img;
    float acy = (av.y + av.w) * 0.5f / img;

    float fs   = 1.0f / (float)(4 << lvl);
    float MINS = 0.01f * (float)(lvl + 1);
    float MAXS = 0.06f + 0.02f * (float)lvl;
    float WSC  = 0.8f + 0.1f * (float)lvl;
    float HSC  = 1.0f + 0.2f * (float)lvl;

    float dx = clipf(d0 * 0.2f, -1.f, 1.f) * fs * aw;
    float dy = clipf(d1 * 0.2f, -1.f, 1.f) * fs * ah;
    float cx = clipf(acx + dx, 0.f, 1.f);
    float cy = clipf(acy + dy, 0.f, 1.f);
    float dw = clipf(d2 * 0.2f, -2.f, 2.f);
    float dh = clipf(d3 * 0.2f, -2.f, 2.f);
    float pw = clipf(aw * WSC * expf(dw * (0.2f + 0.1f * (float)lvl)), MINS * 0.8f, MAXS * 0.9f);
    float ph = clipf(ah * HSC * expf(dh * (0.4f + 0.2f * (float)lvl)), MINS * 0.8f, MAXS * 1.2f);

    float x0 = clipf(cx - 0.5f * pw, 0.f, 1.f);
    float y0 = clipf(cy - 0.5f * ph, 0.f, 1.f);
    float x1 = clipf(cx + 0.5f * pw, 0.f, 1.f);
    float y1 = clipf(cy + 0.5f * ph, 0.f, 1.f);
    float w = x1 - x0, h = y1 - y0;

    bool valid = (ms > 0.15f) && (w > MINS * 0.5f) && (h > MINS * 0.5f)
                              && (w < MAXS * 2.f)  && (h < MAXS * 2.f);
    float key = valid ? ms : -1.0f;

    ws_box[t] = make_float4(x0, y0, x1, y1);
    ws_u[t]   = f2u(key);
    ws_ms[t]  = ms;
    ws_mc[t]  = mc;
}

// ---------- kernel 2: exact top-512 (binary-search select) + sort + NMS ----------

__global__ __launch_bounds__(1024) void select_nms_kernel(
    const float4* __restrict__ ws_box, const unsigned* __restrict__ ws_u,
    const float* __restrict__ ws_ms, const int* __restrict__ ws_mc,
    float* __restrict__ out)
{
    __shared__ unsigned long long sel[KSEL];        // packed (key<<32 | ~idx), sorted desc
    __shared__ __align__(16) float lbox[KSEL * 4];  // async b128 dest: must be 16B aligned
    __shared__ float    lms[KSEL];                  // async b32 gather
    __shared__ int      lmc[KSEL];                  // async b32 gather
    __shared__ unsigned sup[KSEL * 16];             // 512x512 suppression bit-matrix (32KB)
    __shared__ int      keepF[KSEL];
    __shared__ unsigned lred[33];
    __shared__ int      slotCnt;

    const int tid = threadIdx.x;
    const int bx  = blockIdx.x;       // b*4 + lvl
    const int lvl = bx & 3;
    const int b   = bx >> 2;

    const int N = N_LVL0 >> (2 * lvl);     // 49152, 12288, 3072, 768
    int off = 0;
    if (lvl >= 1) off += 49152;
    if (lvl >= 2) off += 12288;
    if (lvl >= 3) off += 3072;
    const int base = b * TOT_ANCH + off;

    // cache this thread's keys in registers (<=48 each)
    unsigned uloc[48];
    #pragma unroll
    for (int p = 0; p < 48; p++) {
        int idx = tid + (p << 10);
        uloc[p] = (idx < N) ? ws_u[base + idx] : 0u;
    }

    // ---- exact 512th-largest key: binary search on monotonic u32 space ----
    unsigned lo = 0u, hi = 0xFFFFFFFFu;
    while (lo < hi) {
        unsigned mid = (unsigned)((((unsigned long long)lo + (unsigned long long)hi) + 1ull) >> 1);
        unsigned c = 0;
        #pragma unroll
        for (int p = 0; p < 48; p++) {
            int idx = tid + (p << 10);
            if (idx < N && uloc[p] >= mid) c++;
        }
        c = blockSum(c, lred, tid);
        if (c >= KSEL) lo = mid; else hi = mid - 1u;
    }
    const unsigned ustar = lo;

    // count strictly greater
    unsigned cg = 0;
    #pragma unroll
    for (int p = 0; p < 48; p++) {
        int idx = tid + (p << 10);
        if (idx < N && uloc[p] > ustar) cg++;
    }
    cg = blockSum(cg, lred, tid);
    const unsigned need = KSEL - cg;      // ties to take, smallest indices first (jax tie order)

    // smallest index threshold Istar: exactly `need` ties have idx < Istar
    unsigned loI = 0u, hiI = (unsigned)N;
    while (loI < hiI) {
        unsigned mid = (loI + hiI) >> 1;
        unsigned c = 0;
        #pragma unroll
        for (int p = 0; p < 48; p++) {
            int idx = tid + (p << 10);
            if ((unsigned)idx < mid && uloc[p] == ustar) c++;
        }
        c = blockSum(c, lred, tid);
        if (c >= need) hiI = mid; else loI = mid + 1u;
    }
    const unsigned Istar = loI;

    // ---- compact the exactly-512 winners ----
    if (tid == 0) slotCnt = 0;
    __syncthreads();
    #pragma unroll
    for (int p = 0; p < 48; p++) {
        int idx = tid + (p << 10);
        if (idx < N) {
            unsigned u = uloc[p];
            if (u > ustar || (u == ustar && (unsigned)idx < Istar)) {
                int s = atomicAdd(&slotCnt, 1);
                sel[s] = ((unsigned long long)u << 32) | (unsigned)(~(unsigned)idx);
            }
        }
    }
    __syncthreads();

    // ---- bitonic sort 512 packed keys, descending (key desc, idx asc) ----
    for (int k = 2; k <= KSEL; k <<= 1) {
        for (int j = k >> 1; j > 0; j >>= 1) {
            __syncthreads();
            if (tid < KSEL) {
                int e = tid, pr = e ^ j;
                if (pr > e) {
                    unsigned long long x = sel[e], y = sel[pr];
                    bool up = ((e & k) == 0);
                    bool sw = up ? (x < y) : (x > y);
                    if (sw) { sel[e] = y; sel[pr] = x; }
                }
            }
        }
    }
    __syncthreads();

    // ---- scattered gather of winners via CDNA5 async global->LDS loads ----
    if (tid < KSEL) {
        unsigned long long pk = sel[tid];
        unsigned idx = ~((unsigned)pk);
        int G = base + (int)idx;
        async_load_b128(lds_off(&lbox[tid * 4]), (const void*)&ws_box[G]);
        async_load_b32(lds_off(&lms[tid]), (const void*)&ws_ms[G]);
        async_load_b32(lds_off(&lmc[tid]), (const void*)&ws_mc[G]);
    }
    wait_asynccnt0();
    __syncthreads();

    // ---- parallel 512x512 suppression bit-matrix: sup[i][w] bit q = IoU(i, 32w+q) > NMS_T ----
    #pragma unroll
    for (int p = 0; p < 8; p++) {
        int id = tid + (p << 10);          // 0..8191
        int i  = id >> 4;
        int w  = id & 15;
        float ax0 = lbox[i * 4 + 0], ay0 = lbox[i * 4 + 1];
        float ax1 = lbox[i * 4 + 2], ay1 = lbox[i * 4 + 3];
        float areaA = (ax1 - ax0) * (ay1 - ay0);
        unsigned bits = 0;
        for (int q = 0; q < 32; q++) {
            int j = (w << 5) + q;
            float bx0 = lbox[j * 4 + 0], by0 = lbox[j * 4 + 1];
            float bx1 = lbox[j * 4 + 2], by1 = lbox[j * 4 + 3];
            float areaB = (bx1 - bx0) * (by1 - by0);
            float ix = fmaxf(fminf(ax1, bx1) - fmaxf(ax0, bx0), 0.f);
            float iy = fmaxf(fminf(ay1, by1) - fmaxf(ay0, by0), 0.f);
            float inter = ix * iy;
            float iou = inter / (areaA + areaB - inter + 1e-9f);
            bits |= (iou > 0.5f ? 1u : 0u) << q;
        }
        sup[i * 16 + w] = bits;
    }
    __syncthreads();

    // ---- greedy sweep: single wave32, barrier-free; lane w holds keep-bits for j in [32w,32w+32) ----
    if (tid < 32) {
        const int lane = tid;
        const unsigned INVALID_U = f2u(-1.0f);
        unsigned keepw = 0;
        for (int i = 0; i < KSEL; i++) {
            unsigned s = (lane < 16) ? (sup[i * 16 + lane] & keepw) : 0u;
            int any = __any((int)(s != 0u));
            unsigned ui = (unsigned)(sel[i] >> 32);
            int kp = (ui != INVALID_U) && (!any);
            if (lane == (i >> 5)) keepw |= ((unsigned)kp) << (i & 31);
            if (lane == 0) keepF[i] = kp;
        }
    }
    __syncthreads();

    // ---- masked output writes: boxes | scores | labels | keep ----
    if (tid < KSEL) {
        int kp = keepF[tid];
        int o = bx * KSEL + tid;                   // (b*4+lvl)*512 + k
        float4 bv;
        bv.x = kp ? lbox[tid * 4 + 0] : 0.f;
        bv.y = kp ? lbox[tid * 4 + 1] : 0.f;
        bv.z = kp ? lbox[tid * 4 + 2] : 0.f;
        bv.w = kp ? lbox[tid * 4 + 3] : 0.f;
        ((float4*)out)[o] = bv;                    // boxes region: 32768 floats
        out[32768 + o] = kp ? lms[tid] : 0.f;      // scores
        out[40960 + o] = kp ? (float)(lmc[tid] + 1) : 0.f;  // labels
        out[49152 + o] = kp ? 1.f : 0.f;           // keep
    }
}

// ---------- host launch ----------

extern "C" void kernel_launch(void* const* d_in, const int* in_sizes, int n_in,
                              void* d_out, int out_size, void* d_ws, size_t ws_size,
                              hipStream_t stream) {
    const float* cls0 = (const float*)d_in[0];
    const float* box0 = (const float*)d_in[1];
    const float* anc0 = (const float*)d_in[2];
    const float* cls1 = (const float*)d_in[3];
    const float* box1 = (const float*)d_in[4];
    const float* anc1 = (const float*)d_in[5];
    const float* cls2 = (const float*)d_in[6];
    const float* box2 = (const float*)d_in[7];
    const float* anc2 = (const float*)d_in[8];
    const float* cls3 = (const float*)d_in[9];
    const float* box3 = (const float*)d_in[10];
    const float* anc3 = (const float*)d_in[11];
    const int*   imgp = (const int*)d_in[12];

    const size_t NA = (size_t)NB * TOT_ANCH;   // 261120 anchors total
    char* ws = (char*)d_ws;
    float4*   ws_box = (float4*)ws;                      // 16B/anchor
    unsigned* ws_u   = (unsigned*)(ws + NA * 16);        // 4B/anchor
    float*    ws_ms  = (float*)(ws + NA * 20);           // 4B/anchor
    int*      ws_mc  = (int*)(ws + NA * 24);             // 4B/anchor (7.3MB total)

    float* out = (float*)d_out;

    decode_kernel<<<dim3((unsigned)((NA + 255) / 256)), dim3(256), 0, stream>>>(
        cls0, box0, anc0, cls1, box1, anc1, cls2, box2, anc2, cls3, box3, anc3,
        imgp, ws_box, ws_u, ws_ms, ws_mc);

    select_nms_kernel<<<dim3(16), dim3(1024), 0, stream>>>(
        ws_box, ws_u, ws_ms, ws_mc, out);

    (void)in_sizes; (void)n_in; (void)out_size; (void)ws_size;
}